// AdaptiveUnivariateFunction_13649406067516
// MI455X (gfx1250) — compile-verified
//
#include <hip/hip_runtime.h>

typedef float f32x4 __attribute__((ext_vector_type(4)));
typedef float f32x2 __attribute__((ext_vector_type(2)));

#define BLOCK  256
#define GRID   2048
#define STAGES 4

// ---------- order-preserving float <-> uint for atomic min/max ----------
__device__ __forceinline__ unsigned f2ord(float f) {
  unsigned u = __float_as_uint(f);
  return (u & 0x80000000u) ? ~u : (u | 0x80000000u);
}
__device__ __forceinline__ float ord2f(unsigned u) {
  unsigned b = (u & 0x80000000u) ? (u & 0x7FFFFFFFu) : ~u;
  return __uint_as_float(b);
}

__global__ void init_ws_kernel(unsigned* ws) {
  ws[0] = 0xFFFFFFFFu;  // min accumulator (ordered-uint +inf)
  ws[1] = 0x00000000u;  // max accumulator (ordered-uint -inf)
}

// ---------- pass 1: global min/max reduction (memory bound) ----------
__global__ __launch_bounds__(BLOCK) void minmax_kernel(const float* __restrict__ x,
                                                       long long n,
                                                       unsigned* __restrict__ ws) {
  const long long gid    = (long long)blockIdx.x * blockDim.x + threadIdx.x;
  const long long stride = (long long)gridDim.x * blockDim.x;
  const long long n4     = n >> 2;
  const f32x4* __restrict__ x4 = (const f32x4*)x;

  float mn = __builtin_inff();
  float mx = -__builtin_inff();
  for (long long i = gid; i < n4; i += stride) {
    long long pf = i + 8 * stride;
    if (pf < n4) __builtin_prefetch((const void*)&x4[pf], 0, 0);  // global_prefetch_b8
    f32x4 v = x4[i];
    mn = fminf(mn, fminf(fminf(v.x, v.y), fminf(v.z, v.w)));
    mx = fmaxf(mx, fmaxf(fmaxf(v.x, v.y), fmaxf(v.z, v.w)));
  }
  const long long t0 = n4 << 2;
  if (gid < (n - t0)) { float v = x[t0 + gid]; mn = fminf(mn, v); mx = fmaxf(mx, v); }

  // wave32 butterfly reduction
  for (int off = 16; off > 0; off >>= 1) {
    mn = fminf(mn, __shfl_xor(mn, off, 32));
    mx = fmaxf(mx, __shfl_xor(mx, off, 32));
  }
  if ((threadIdx.x & 31) == 0) {
    atomicMin(&ws[0], f2ord(mn));
    atomicMax(&ws[1], f2ord(mx));
  }
}

// ---------- CDNA5 async global->LDS staging ----------
#define WAIT_ASYNC(N) asm volatile("s_wait_asynccnt " #N ::: "memory")

__device__ __forceinline__ void async_load16(unsigned lds_off, const float* base,
                                             unsigned byte_off) {
  // GVS form: vdst = LDS byte address, vaddr = 32-bit offset, saddr = 64-bit base
  asm volatile("global_load_async_to_lds_b128 %0, %1, %2"
               :
               : "v"(lds_off), "v"(byte_off), "s"(base)
               : "memory");
}

// ---------- spline: idx = floor(31*xn); t = frac; lerp(cp[idx], cp[idx+1]) ----------
__device__ __forceinline__ float eval1(float xv, float scale31, float bias,
                                       const f32x2* __restrict__ table) {
  float f = fmaf(xv, scale31, bias);   // = 31 * (x - min) / (max - min + 1e-6) in [0,31)
  int idx = (int)f;                    // trunc == floor (f >= ~0; tiny negatives -> 0)
  idx = idx > 30 ? 30 : idx;
  f32x2 q = table[idx];                // ds_load_b64: {cp[idx], cp[idx+1]}
  float t = f - (float)idx;            // uniform knots: 31*knots[idx] == idx
  return fmaf(t, q.y - q.x, q.x);      // cp0 + t*(cp1 - cp0)
}

__global__ __launch_bounds__(BLOCK) void spline_kernel(const float* __restrict__ x,
                                                       const float* __restrict__ cp,
                                                       float* __restrict__ out,
                                                       long long n,
                                                       const unsigned* __restrict__ ws) {
  __shared__ f32x4 stage[STAGES * BLOCK];  // 16 KB: per-thread 16B slots, 4 pipeline stages
  __shared__ f32x2 table[32];              // lerp lookup table (pairs)

  const int tid = threadIdx.x;
  if (tid < 31) {
    f32x2 e;
    e.x = cp[tid];
    e.y = cp[tid + 1];
    table[tid] = e;
  }
  __syncthreads();

  const float mn      = ord2f(ws[0]);
  const float mx      = ord2f(ws[1]);
  const float inv     = 1.0f / (mx - mn + 1e-6f);
  const float scale31 = inv * 31.0f;
  const float bias    = -mn * scale31;

  const long long tpg   = (long long)gridDim.x * blockDim.x;
  const long long gid0  = (long long)blockIdx.x * blockDim.x + tid;
  const long long n4    = n >> 2;
  const long long iters = n4 / tpg;  // uniform across all threads/waves
  f32x4* __restrict__ out4 = (f32x4*)out;

  const unsigned lds0 = (unsigned)(unsigned long long)&stage[tid];

  // Traverse stripes in REVERSE: pass 1 finished with the tail stripes hottest in L2
  // (192 MB L2 vs 256 MB array), so read those first while they're still resident.
  // row(k) = iters-1-k; async pipeline depth/order is unchanged.

  // prologue: fill the async pipeline
  const long long pro = iters < STAGES ? iters : STAGES;
  for (long long p = 0; p < pro; ++p)
    async_load16(lds0 + (unsigned)p * (BLOCK * 16u), x,
                 (unsigned)((gid0 + (iters - 1 - p) * tpg) << 4));

  for (long long k = 0; k < iters; ++k) {
    const int s = (int)(k & (STAGES - 1));
    const long long rem = iters - 1 - k;  // async loads still in flight after load k
    if (rem >= 3)      WAIT_ASYNC(3);     // guarantees load k has landed in LDS
    else if (rem == 2) WAIT_ASYNC(2);
    else if (rem == 1) WAIT_ASYNC(1);
    else               WAIT_ASYNC(0);

    f32x4 v = stage[s * BLOCK + tid];             // ds_load_b128 from own slot
    asm volatile("s_wait_dscnt 0" ::: "memory");  // read done before slot reuse

    const long long nk = k + STAGES;              // refill this slot
    if (nk < iters)
      async_load16(lds0 + (unsigned)s * (BLOCK * 16u), x,
                   (unsigned)((gid0 + (iters - 1 - nk) * tpg) << 4));

    f32x4 r;
    r.x = eval1(v.x, scale31, bias, table);
    r.y = eval1(v.y, scale31, bias, table);
    r.z = eval1(v.z, scale31, bias, table);
    r.w = eval1(v.w, scale31, bias, table);
    __builtin_nontemporal_store(r, &out4[gid0 + (iters - 1 - k) * tpg]);
  }

  // float4 tail (n4 % tpg) — plain loads, at most one iteration
  for (long long i = iters * tpg + gid0; i < n4; i += tpg) {
    f32x4 v = ((const f32x4*)x)[i];
    f32x4 r;
    r.x = eval1(v.x, scale31, bias, table);
    r.y = eval1(v.y, scale31, bias, table);
    r.z = eval1(v.z, scale31, bias, table);
    r.w = eval1(v.w, scale31, bias, table);
    out4[i] = r;
  }
  // scalar tail (n % 4)
  const long long t0 = n4 << 2;
  if (gid0 < n - t0) out[t0 + gid0] = eval1(x[t0 + gid0], scale31, bias, table);
}

extern "C" void kernel_launch(void* const* d_in, const int* in_sizes, int n_in,
                              void* d_out, int out_size, void* d_ws, size_t ws_size,
                              hipStream_t stream) {
  const float* x  = (const float*)d_in[0];
  const float* cp = (const float*)d_in[1];
  float* out      = (float*)d_out;
  unsigned* ws    = (unsigned*)d_ws;
  const long long n = (long long)out_size;  // == in_sizes[0] == 64 * 1048576

  init_ws_kernel<<<1, 1, 0, stream>>>(ws);
  minmax_kernel<<<GRID, BLOCK, 0, stream>>>(x, n, ws);
  spline_kernel<<<GRID, BLOCK, 0, stream>>>(x, cp, out, n, ws);
}